// TreeTripletLoss_19722489823690
// MI455X (gfx1250) — compile-verified
//
#include <hip/hip_runtime.h>

typedef float v2f __attribute__((ext_vector_type(2)));
typedef float v8f __attribute__((ext_vector_type(8)));

#define NUMC   19
#define MAXT   200
#define MARGIN 0.6f

// Fixed problem geometry from the reference setup.
constexpr int C_  = 256;
constexpr int HW_ = 128 * 128;          // H*W
// pixel n -> feats element for channel c: ((n>>14)*C + c)*HW + (n & (HW-1))

__device__ __forceinline__ float pick8(v8f v, int j) {
    float r = v[0];
    r = (j == 1) ? v[1] : r;
    r = (j == 2) ? v[2] : r;
    r = (j == 3) ? v[3] : r;
    r = (j == 4) ? v[4] : r;
    r = (j == 5) ? v[5] : r;
    r = (j == 6) ? v[6] : r;
    r = (j == 7) ? v[7] : r;
    return r;
}

// ---------------- Stage 1: full per-class counts (LDS histogram) ----------------
__global__ void k_count(const int* __restrict__ lab, int N, int* __restrict__ counts) {
    __shared__ int h[32];
    if (threadIdx.x < 32) h[threadIdx.x] = 0;
    __syncthreads();
    int i = blockIdx.x * blockDim.x + threadIdx.x;
    int stride = gridDim.x * blockDim.x;
    for (; i < N; i += stride) {
        int c = lab[i] & 31;
        atomicAdd(&h[c], 1);
    }
    __syncthreads();
    if (threadIdx.x < 32) {
        int v = h[threadIdx.x];
        if (v) atomicAdd(&counts[threadIdx.x], v);
    }
}

// ---------------- Stage 2: ordered first-200 index selection per class ----------
// Single wave32; ballot + prefix-popc gives exact in-order slots (jnp.nonzero order).
__global__ void k_select(const int* __restrict__ lab, int N, int* __restrict__ sel_a) {
    const int lane = threadIdx.x;   // 0..31, one wave
    int base[NUMC];
    #pragma unroll
    for (int c = 0; c < NUMC; ++c) base[c] = 0;

    for (int chunk = 0; chunk < N; chunk += 32) {
        bool all_done = true;
        #pragma unroll
        for (int c = 1; c < NUMC; ++c) all_done = all_done && (base[c] >= MAXT);
        if (all_done) break;                       // wave-uniform

        int idx = chunk + lane;
        int c = (idx < N) ? lab[idx] : -1;

        for (int cls = 1; cls < NUMC; ++cls) {
            if (base[cls] >= MAXT) continue;       // wave-uniform
            unsigned m = __builtin_amdgcn_ballot_w32(c == cls);
            if (m) {                               // wave-uniform (sgpr)
                unsigned below = m & ((1u << lane) - 1u);
                int slot = base[cls] + __popc(below);
                if ((c == cls) && slot < MAXT)
                    sel_a[(cls - 1) * MAXT + slot] = idx;
                base[cls] += __popc(m);
            }
        }
    }
}

// ---------------- Stage 3: k-way merge -> positive / negative lists -------------
__global__ void k_merge(const int* __restrict__ counts,
                        const int* __restrict__ sel_a,
                        int* __restrict__ sel_p,
                        int* __restrict__ sel_n) {
    int t = threadIdx.x;
    if (t >= 20) return;
    const int* lists[10];
    int len[10], ptr[10];
    int k = 0;
    int* out;
    if (t < 18) {                       // positive list for class t+1
        int cls = t + 1;
        int lo = (cls <= 10) ? 1 : 11;
        int hi = (cls <= 10) ? 10 : 18;
        for (int c = lo; c <= hi; ++c) {
            if (c == cls) continue;
            lists[k] = sel_a + (c - 1) * MAXT;
            len[k] = min(counts[c], MAXT);
            ptr[k] = 0;
            ++k;
        }
        out = sel_p + t * MAXT;
    } else {                            // t==18: lower union (neg for upper anchors); t==19: upper union
        int lo = (t == 18) ? 11 : 1;
        int hi = (t == 18) ? 18 : 10;
        for (int c = lo; c <= hi; ++c) {
            lists[k] = sel_a + (c - 1) * MAXT;
            len[k] = min(counts[c], MAXT);
            ptr[k] = 0;
            ++k;
        }
        out = sel_n + (t - 18) * MAXT;
    }
    for (int o = 0; o < MAXT; ++o) {
        int best = 0x7FFFFFFF, bi = -1;
        for (int j = 0; j < k; ++j) {
            if (ptr[j] < len[j]) {
                int v = lists[j][ptr[j]];
                if (v < best) { best = v; bi = j; }
            }
        }
        if (bi >= 0) { out[o] = best; ptr[bi]++; }
        else         { out[o] = 0; }        // matches jnp.nonzero fill_value=0
    }
}

// ---------------- Stage 4: WMMA triplet dot products ----------------------------
// One wave per (class, tile of 16 triplets). f32 WMMA keeps reference precision.
__global__ void k_triplet(const float* __restrict__ feats,
                          const int* __restrict__ counts,
                          const int* __restrict__ sel_a,
                          const int* __restrict__ sel_p,
                          const int* __restrict__ sel_n,
                          float* __restrict__ classSum) {
    const int clsIdx = blockIdx.x;      // 0..17
    const int cls = clsIdx + 1;
    const int tile = blockIdx.y;        // 0..12 (13*16 = 208 >= 200)

    int upper = 0, lower = 0;
    #pragma unroll
    for (int c = 1; c <= 10; ++c) upper += counts[c];
    #pragma unroll
    for (int c = 11; c <= 18; ++c) lower += counts[c];
    int n_a = counts[cls];
    int n_p, n_n;
    if (cls <= 10) { n_p = upper - n_a; n_n = lower; }
    else           { n_p = lower - n_a; n_n = upper; }
    int m = min(min(n_a, n_p), min(n_n, MAXT));
    int t0 = tile * 16;
    if (t0 >= m) return;                // wave-uniform: EXEC stays all-ones for WMMA

    const int lane = threadIdx.x;       // one wave32
    const int M  = lane & 15;           // row (and column) index within the tile
    const int Kb = (lane >> 4) * 2;     // K-pair base per ISA 16x4 / 4x16 f32 layout
    int t  = t0 + M;
    int tc = min(t, MAXT - 1);          // clamp tail (masked out below anyway)

    int ra = sel_a[clsIdx * MAXT + tc];
    int rp = sel_p[clsIdx * MAXT + tc];
    int rn = sel_n[((cls <= 10) ? 0 : 1) * MAXT + tc];

    auto rowbase = [](int n) -> size_t {
        return ((size_t)(n >> 14) * (size_t)C_) * (size_t)HW_ + (size_t)(n & (HW_ - 1));
    };
    const size_t ba = rowbase(ra), bp = rowbase(rp), bn = rowbase(rn);
    const size_t cs = (size_t)HW_;      // channel stride in floats

    v8f accN = {};
    v8f accP = {};
    #pragma unroll 4
    for (int k4 = 0; k4 < C_ / 4; ++k4) {
        const int c0 = k4 * 4 + Kb;
        v2f a, bN, bP;
        a.x  = feats[ba + (size_t)c0 * cs];
        a.y  = feats[ba + (size_t)(c0 + 1) * cs];
        bN.x = feats[bn + (size_t)c0 * cs];
        bN.y = feats[bn + (size_t)(c0 + 1) * cs];
        bP.x = feats[bp + (size_t)c0 * cs];
        bP.y = feats[bp + (size_t)(c0 + 1) * cs];
        accN = __builtin_amdgcn_wmma_f32_16x16x4_f32(false, a, false, bN, (short)0, accN, false, false);
        accP = __builtin_amdgcn_wmma_f32_16x16x4_f32(false, a, false, bP, (short)0, accP, false, false);
    }

    // Diagonal of 16x16 f32 C/D layout: i in 0..7 -> lane i, comp i;
    // i in 8..15 -> lane i+16 (24..31), comp lane-24.
    const bool have = (lane < 8) || (lane >= 24);
    const int  comp = (lane < 8) ? lane : (lane - 24);
    const int  ti   = (lane < 8) ? (t0 + lane) : (t0 + (lane - 16));
    float dn = pick8(accN, comp);
    float dp = pick8(accP, comp);
    float tl = fmaxf(dn - dp + MARGIN, 0.0f);
    if (have && ti < m) atomicAdd(&classSum[clsIdx], tl);
}

// ---------------- Stage 5: finalize ---------------------------------------------
__global__ void k_final(const int* __restrict__ counts,
                        const float* __restrict__ classSum,
                        float* __restrict__ out) {
    if (threadIdx.x != 0 || blockIdx.x != 0) return;
    int upper = 0, lower = 0;
    for (int c = 1; c <= 10; ++c) upper += counts[c];
    for (int c = 11; c <= 18; ++c) lower += counts[c];
    float total = 0.0f;
    int cnt = 0;
    for (int cls = 1; cls <= 18; ++cls) {
        int n_a = counts[cls], n_p, n_n;
        if (cls <= 10) { n_p = upper - n_a; n_n = lower; }
        else           { n_p = lower - n_a; n_n = upper; }
        int m = min(min(n_a, n_p), min(n_n, MAXT));
        if (m > 0) { total += classSum[cls - 1] / (float)m; ++cnt; }
    }
    out[0] = (cnt > 0) ? total / (float)cnt : 0.0f;
    out[1] = (float)cnt;
}

extern "C" void kernel_launch(void* const* d_in, const int* in_sizes, int n_in,
                              void* d_out, int out_size, void* d_ws, size_t ws_size,
                              hipStream_t stream) {
    const float* feats  = (const float*)d_in[0];
    const int*   labels = (const int*)d_in[1];
    const int N = in_sizes[1];          // 131072

    // Workspace layout (ints unless noted):
    int* counts = (int*)d_ws;                    // [32]
    int* sel_a  = counts + 32;                   // [18*200]
    int* sel_p  = sel_a + 18 * MAXT;             // [18*200]
    int* sel_n  = sel_p + 18 * MAXT;             // [2*200]
    float* classSum = (float*)(sel_n + 2 * MAXT);// [18] floats

    size_t zbytes = (size_t)(32 + 18 * MAXT + 18 * MAXT + 2 * MAXT) * sizeof(int)
                  + 18 * sizeof(float);
    hipMemsetAsync(d_ws, 0, zbytes, stream);     // zero counters + pad lists with 0

    k_count  <<<dim3(256),     dim3(256), 0, stream>>>(labels, N, counts);
    k_select <<<dim3(1),       dim3(32),  0, stream>>>(labels, N, sel_a);
    k_merge  <<<dim3(1),       dim3(32),  0, stream>>>(counts, sel_a, sel_p, sel_n);
    k_triplet<<<dim3(18, 13),  dim3(32),  0, stream>>>(feats, counts, sel_a, sel_p, sel_n, classSum);
    k_final  <<<dim3(1),       dim3(1),   0, stream>>>(counts, classSum, (float*)d_out);
}